// GNNRegressor_44195213476076
// MI455X (gfx1250) — compile-verified
//
#include <hip/hip_runtime.h>

typedef float v2f __attribute__((ext_vector_type(2)));
typedef float v8f __attribute__((ext_vector_type(8)));

// ---------------- utility kernels ----------------

__global__ void k_zero(float* __restrict__ p, long long n) {
  long long i = (long long)blockIdx.x * blockDim.x + threadIdx.x;
  if (i < n) p[i] = 0.0f;
}

// degree accumulation + int64 -> int32 edge index conversion (done once)
__global__ void k_prep_edges(const long long* __restrict__ ei, const float* __restrict__ ew,
                             int* __restrict__ row32, int* __restrict__ col32,
                             float* __restrict__ deg, int E) {
  int e = blockIdx.x * blockDim.x + threadIdx.x;
  if (e >= E) return;
  int r = (int)ei[e];
  int c = (int)ei[(long long)E + e];
  row32[e] = r;
  col32[e] = c;
  atomicAdd(&deg[c], ew[e]);
}

// dinv = rsqrt(deg + 1)   (self-loop weight 1.0; deg+1 >= 1 so always > 0)
__global__ void k_dinv(float* __restrict__ deg_dinv, int N) {
  int n = blockIdx.x * blockDim.x + threadIdx.x;
  if (n >= N) return;
  deg_dinv[n] = rsqrtf(deg_dinv[n] + 1.0f);
}

// norm[e] = dinv[row]*ew*dinv[col]  (computed once, reused by all 3 layers)
__global__ void k_norm(const int* __restrict__ row32, const int* __restrict__ col32,
                       const float* __restrict__ ew, const float* __restrict__ dinv,
                       float* __restrict__ norm, int E) {
  int e = blockIdx.x * blockDim.x + threadIdx.x;
  if (e >= E) return;
  norm[e] = dinv[row32[e]] * ew[e] * dinv[col32[e]];
}

// ---------------- WMMA GEMM: out[N,64] = in[N,64] @ W[64,64] (fp32) ----------------
// One wave per 16x16 output tile; 4 waves/block cover the 64 output columns.
// V_WMMA_F32_16X16X4_F32, K stepped by 4, 16 WMMAs per tile.
__global__ __launch_bounds__(128) void k_gemm64(const float* __restrict__ in,
                                                const float* __restrict__ W,
                                                float* __restrict__ out, int nrows) {
  const int lane = threadIdx.x & 31;
  const int wave = threadIdx.x >> 5;        // n-tile 0..3
  const int m0   = blockIdx.x * 16;         // 16-row tile base
  const int ml   = lane & 15;
  const int half = lane >> 4;               // 0 or 1
  const int n    = wave * 16 + ml;          // output column

  int arow_i = m0 + ml;
  if (arow_i >= nrows) arow_i = nrows - 1;  // clamp reads; stores guarded below
  const float* arow = in + (long long)arow_i * 64;

  v8f acc = {};
#pragma unroll
  for (int kb = 0; kb < 64; kb += 4) {
    const int k2 = kb + 2 * half;           // ISA A/B layout: VGPR r, lane-half h -> K = r + 2h
    v2f a, b;
    a.x = arow[k2 + 0];
    a.y = arow[k2 + 1];
    b.x = W[(k2 + 0) * 64 + n];
    b.y = W[(k2 + 1) * 64 + n];
    acc = __builtin_amdgcn_wmma_f32_16x16x4_f32(
        /*neg_a=*/false, a, /*neg_b=*/false, b,
        /*c_mod=*/(short)0, acc, /*reuse_a=*/false, /*reuse_b=*/false);
  }

  // C/D layout: VGPR r, lane-half h -> row m0 + r + 8h, col = n
#pragma unroll
  for (int r = 0; r < 8; ++r) {
    const int orow = m0 + r + 8 * half;
    if (orow < nrows) out[(long long)orow * 64 + n] = acc[r];
  }
}

// ---------------- edge scatter: agg[col] += h[row] * norm ----------------
// thread = (edge, feature); 64 consecutive lanes read/atomically write one 256B row
__global__ void k_scatter(const float* __restrict__ h, const int* __restrict__ row32,
                          const int* __restrict__ col32, const float* __restrict__ norm,
                          float* __restrict__ agg, int E) {
  long long tid = (long long)blockIdx.x * blockDim.x + threadIdx.x;
  if (tid >= (long long)E * 64) return;
  const int e = (int)(tid >> 6);
  const int f = (int)(tid & 63);
  const float v = h[(long long)row32[e] * 64 + f] * norm[e];
  atomicAdd(&agg[(long long)col32[e] * 64 + f], v);
}

// out = agg + h*dinv^2 + b  (+ optional ReLU); safe to write in-place over agg
__global__ void k_combine(const float* __restrict__ agg, const float* __restrict__ h,
                          const float* __restrict__ dinv, const float* __restrict__ b,
                          float* __restrict__ out, int N, int do_relu) {
  long long tid = (long long)blockIdx.x * blockDim.x + threadIdx.x;
  if (tid >= (long long)N * 64) return;
  const int n = (int)(tid >> 6);
  const int f = (int)(tid & 63);
  const float d = dinv[n];
  float v = agg[tid] + h[tid] * d * d + b[f];
  if (do_relu) v = fmaxf(v, 0.0f);
  out[tid] = v;
}

// ---------------- global mean pool ----------------
__global__ void k_pool_cnt(const long long* __restrict__ batch, float* __restrict__ cnts, int N) {
  int n = blockIdx.x * blockDim.x + threadIdx.x;
  if (n >= N) return;
  atomicAdd(&cnts[(int)batch[n]], 1.0f);
}

__global__ void k_pool_sum(const float* __restrict__ h, const long long* __restrict__ batch,
                           float* __restrict__ sums, int N) {
  long long tid = (long long)blockIdx.x * blockDim.x + threadIdx.x;
  if (tid >= (long long)N * 64) return;
  const int n = (int)(tid >> 6);
  const int f = (int)(tid & 63);
  atomicAdd(&sums[(int)batch[n] * 64 + f], h[tid]);
}

// ---------------- regression head (tiny): one thread per graph ----------------
__global__ void k_head(const float* __restrict__ sums, const float* __restrict__ cnts,
                       const float* __restrict__ fc1w, const float* __restrict__ fc1b,
                       const float* __restrict__ fc2w, const float* __restrict__ fc2b,
                       const float* __restrict__ ow, const float* __restrict__ ob,
                       float* __restrict__ out, int G) {
  int g = blockIdx.x * blockDim.x + threadIdx.x;
  if (g >= G) return;
  const float inv_c = 1.0f / fmaxf(cnts[g], 1.0f);
  float gm[64];
#pragma unroll
  for (int i = 0; i < 64; ++i) gm[i] = sums[g * 64 + i] * inv_c;
  float s = fc2b[0];
  for (int j = 0; j < 32; ++j) {
    float a = fc1b[j];
#pragma unroll
    for (int i = 0; i < 64; ++i) a += gm[i] * fc1w[i * 32 + j];
    a = fmaxf(a, 0.0f);
    s += a * fc2w[j];
  }
  out[g] = s * ow[0] + ob[0];
}

// ---------------- host orchestration ----------------
extern "C" void kernel_launch(void* const* d_in, const int* in_sizes, int n_in,
                              void* d_out, int out_size, void* d_ws, size_t ws_size,
                              hipStream_t stream) {
  const float*     x     = (const float*)d_in[0];
  const long long* ei    = (const long long*)d_in[1];   // int64 [2,E]
  const float*     ea    = (const float*)d_in[2];       // [E,1]
  const long long* batch = (const long long*)d_in[3];   // int64 [N]
  const float* W0 = (const float*)d_in[4];  const float* b0 = (const float*)d_in[5];
  const float* W1 = (const float*)d_in[6];  const float* b1 = (const float*)d_in[7];
  const float* W2 = (const float*)d_in[8];  const float* b2 = (const float*)d_in[9];
  const float* fc1w = (const float*)d_in[10]; const float* fc1b = (const float*)d_in[11];
  const float* fc2w = (const float*)d_in[12]; const float* fc2b = (const float*)d_in[13];
  const float* ow = (const float*)d_in[14]; const float* ob = (const float*)d_in[15];
  float* out = (float*)d_out;

  const int N = in_sizes[0] / 64;
  const int E = in_sizes[2];      // edge_attr has E*1 elements
  const int G = out_size;
  const long long NH = (long long)N * 64;
  const long long EH = (long long)E * 64;

  // workspace carve-out (aligned to 256B)
  char* ws = (char*)d_ws;
  size_t off = 0;
  auto carve = [&](size_t bytes) -> void* {
    void* p = ws + off;
    off = (off + bytes + 255) & ~(size_t)255;
    return p;
  };
  float* bufA  = (float*)carve(NH * sizeof(float));   // GEMM output h
  float* bufB  = (float*)carve(NH * sizeof(float));   // agg / layer out (ping)
  float* bufC  = (float*)carve(NH * sizeof(float));   // agg / layer out (pong)
  float* dinv  = (float*)carve((size_t)N * sizeof(float));
  float* normb = (float*)carve((size_t)E * sizeof(float));
  int*   row32 = (int*)  carve((size_t)E * sizeof(int));
  int*   col32 = (int*)  carve((size_t)E * sizeof(int));
  float* sums  = (float*)carve((size_t)G * 64 * sizeof(float));
  float* cnts  = (float*)carve((size_t)G * sizeof(float));
  (void)ws_size; (void)n_in;

  const int T = 256;

  // --- one-time edge preprocessing ---
  k_zero<<<(int)((N + T - 1) / T), T, 0, stream>>>(dinv, N);                       // deg accumulator
  k_prep_edges<<<(E + T - 1) / T, T, 0, stream>>>(ei, ea, row32, col32, dinv, E);
  k_dinv<<<(N + T - 1) / T, T, 0, stream>>>(dinv, N);
  k_norm<<<(E + T - 1) / T, T, 0, stream>>>(row32, col32, ea, dinv, normb, E);

  // --- GCN layer runner: h = in@W (WMMA); agg = scatter(norm*h[row] -> col); out = agg + h*dinv^2 + b ---
  auto layer = [&](const float* lin, const float* W, const float* b, float* aggb, int relu) {
    k_gemm64<<<(N + 15) / 16, 128, 0, stream>>>(lin, W, bufA, N);
    k_zero<<<(int)((NH + T - 1) / T), T, 0, stream>>>(aggb, NH);
    k_scatter<<<(int)((EH + T - 1) / T), T, 0, stream>>>(bufA, row32, col32, normb, aggb, E);
    k_combine<<<(int)((NH + T - 1) / T), T, 0, stream>>>(aggb, bufA, dinv, b, aggb, N, relu);
  };

  layer(x,    W0, b0, bufB, 1);   // layer 0 -> bufB (ReLU)
  layer(bufB, W1, b1, bufC, 1);   // layer 1 -> bufC (ReLU)
  layer(bufC, W2, b2, bufB, 0);   // layer 2 -> bufB (no ReLU)

  // --- global mean pool + head ---
  k_zero<<<(G * 64 + T - 1) / T, T, 0, stream>>>(sums, (long long)G * 64);
  k_zero<<<(G + T - 1) / T, T, 0, stream>>>(cnts, G);
  k_pool_cnt<<<(N + T - 1) / T, T, 0, stream>>>(batch, cnts, N);
  k_pool_sum<<<(int)((NH + T - 1) / T), T, 0, stream>>>(bufB, batch, sums, N);
  k_head<<<1, 64, 0, stream>>>(sums, cnts, fc1w, fc1b, fc2w, fc2b, ow, ob, out, G);
}